// Ridge_51178830299316
// MI455X (gfx1250) — compile-verified
//
#include <hip/hip_runtime.h>
#include <stdint.h>

// ---------------------------------------------------------------------------
// Block recursive least squares (Sherman-Morrison / Woodbury, J=16 blocking)
// One workgroup per batch. P = (I + X^T X)^-1 lives in LDS (64KB of 320KB).
// All O(D^2) GEMMs run on V_WMMA_F32_16X16X4_F32 tiles in f32.
// X blocks are double-buffered via GLOBAL_LOAD_ASYNC_TO_LDS_B128 (ASYNCcnt)
// so global latency is hidden under the sequential block-step chain.
// ---------------------------------------------------------------------------

typedef float v2f __attribute__((ext_vector_type(2)));
typedef float v8f __attribute__((ext_vector_type(8)));

static constexpr int KD       = 128;        // feature dim
static constexpr int NPTS     = 256;        // sequence length
static constexpr int JB       = 16;         // RLS block size (= WMMA tile)
static constexpr int NBLK     = NPTS / JB;  // 16 sequential block-steps
static constexpr int NWAVES   = 8;
static constexpr int NTHREADS = NWAVES * 32;

__device__ __forceinline__ v8f wmma4(v2f a, v2f b, v8f c) {
  // (neg_a, A, neg_b, B, c_mod, C, reuse_a, reuse_b)
  return __builtin_amdgcn_wmma_f32_16x16x4_f32(false, a, false, b, (short)0, c,
                                               false, false);
}

// Async global->LDS copy of one 16-byte chunk per lane (tracked by ASYNCcnt).
__device__ __forceinline__ void async_b128(uint32_t lds_byte, const float* gsrc) {
  uint64_t ga = (uint64_t)(uintptr_t)gsrc;
  asm volatile("global_load_async_to_lds_b128 %0, %1, off"
               :: "v"(lds_byte), "v"(ga)
               : "memory");
}

__device__ __forceinline__ void wait_asynccnt0() {
  asm volatile("s_wait_asynccnt 0x0" ::: "memory");
}

__global__ __launch_bounds__(NTHREADS)
void rls_block_kernel(const float* __restrict__ data,
                      const float* __restrict__ targets,
                      float* __restrict__ out) {
  const int b    = blockIdx.x;
  const int tid  = threadIdx.x;
  const int lane = tid & 31;
  const int wave = tid >> 5;
  const int half = lane >> 4;   // 0: lanes 0-15, 1: lanes 16-31
  const int l    = lane & 15;

  __shared__ float Ps[KD][KD];     // covariance inverse P (64 KB)
  __shared__ float Xb[2][JB][KD];  // double-buffered X block; cur reused as V
  __shared__ float Ts[JB][KD];     // T = X * P
  __shared__ float ws[KD];         // weights
  __shared__ float Gs[JB][JB];     // G = X P X^T
  __shared__ float Bs[JB][JB];     // in-block gain combinations b_r
  __shared__ float Cs[JB][JB];     // C = sum b b^T / s
  __shared__ float cs[JB];         // c = X w
  __shared__ float ys[JB];         // targets of block
  __shared__ float qs[JB];         // q = sum (e/s) b
  __shared__ float ss[JB];         // s_r

  const float* dptr = data    + (size_t)b * NPTS * KD;
  const float* tptr = targets + (size_t)b * NPTS;
  float*       optr = out     + (size_t)b * NPTS;

  // ---- init: P = (1/lambda) I = I, w = 0 ----
  for (int idx = tid; idx < KD * KD; idx += NTHREADS) {
    int r = idx >> 7, c = idx & (KD - 1);
    Ps[r][c] = (r == c) ? 1.0f : 0.0f;
  }
  if (tid < KD) ws[tid] = 0.0f;

  // ---- prologue: async-load block 0 into Xb[0] ----
  {
    const uint32_t xbase = (uint32_t)(uintptr_t)(&Xb[0][0][0]);
#pragma unroll
    for (int i = 0; i < (JB * KD) / (4 * NTHREADS); ++i) {
      const int e = tid + i * NTHREADS;          // 16-byte chunk index
      async_b128(xbase + (uint32_t)e * 16u, dptr + (size_t)e * 4);
    }
  }

  for (int blk = 0; blk < NBLK; ++blk) {
    const int   base = blk * JB;
    const int   cur  = blk & 1;
    float (*Xs)[KD]  = Xb[cur];

    // current X block's async copy (issued last iteration / prologue) done
    wait_asynccnt0();
    if (tid < JB) ys[tid] = tptr[base + tid];
    __syncthreads();

    // kick off prefetch of the NEXT block into the other buffer
    if (blk + 1 < NBLK) {
      const uint32_t xbase = (uint32_t)(uintptr_t)(&Xb[1 - cur][0][0]);
      const float*   src   = dptr + (size_t)(base + JB) * KD;
#pragma unroll
      for (int i = 0; i < (JB * KD) / (4 * NTHREADS); ++i) {
        const int e = tid + i * NTHREADS;
        async_b128(xbase + (uint32_t)e * 16u, src + (size_t)e * 4);
      }
    }

    // ---- T = X @ P : wave w owns output columns [16w, 16w+16) ----
    {
      const int n0 = wave * 16;
      v8f acc = {};
#pragma unroll 4
      for (int k = 0; k < KD; k += 4) {
        v2f a, bf;
        a.x  = Xs[l][k + 2 * half + 0];
        a.y  = Xs[l][k + 2 * half + 1];
        bf.x = Ps[k + 2 * half + 0][n0 + l];
        bf.y = Ps[k + 2 * half + 1][n0 + l];
        acc = wmma4(a, bf, acc);
      }
#pragma unroll
      for (int r = 0; r < 8; ++r) Ts[r + 8 * half][n0 + l] = acc[r];
    }
    __syncthreads();

    // ---- G = T @ X^T (wave 0) ; c = X @ w (wave 1) ----
    if (wave == 0) {
      v8f acc = {};
#pragma unroll 4
      for (int k = 0; k < KD; k += 4) {
        v2f a, bf;
        a.x  = Ts[l][k + 2 * half + 0];
        a.y  = Ts[l][k + 2 * half + 1];
        bf.x = Xs[l][k + 2 * half + 0];   // B[kk][n] = X[n][k+kk], n = l
        bf.y = Xs[l][k + 2 * half + 1];
        acc = wmma4(a, bf, acc);
      }
#pragma unroll
      for (int r = 0; r < 8; ++r) Gs[r + 8 * half][l] = acc[r];
    } else if (wave == 1) {
      if (lane < JB) {
        float s = 0.0f;
        for (int d = 0; d < KD; ++d) s += Xs[lane][d] * ws[d];
        cs[lane] = s;
      }
    }
    __syncthreads();

    // ---- in-block sequential recursion (wave 0, 16-dim space) ----
    if (wave == 0) {
      const int  m      = lane & 15;
      const bool active = lane < 16;
      float q = 0.0f;
      for (int r = 0; r < JB; ++r) {
        float g  = Gs[r][m];
        // pred_r = c_r + g . q
        float pq = active ? g * q : 0.0f;
        for (int off = 16; off > 0; off >>= 1) pq += __shfl_xor(pq, off, 32);
        float pred = cs[r] + pq;
        if (lane == 0) optr[base + r] = pred;   // pred[0] == 0 naturally
        float e = ys[r] - pred;
        // h_i = b_i . g_r (lane i), alpha_i = h_i / s_i  (valid for i < r)
        float h = 0.0f;
        for (int mm = 0; mm < JB; ++mm) h += Bs[m][mm] * Gs[r][mm];
        float alpha = h / ss[m];
        // b_r = unit_r - sum_{i<r} alpha_i * b_i
        float bm = (m == r) ? 1.0f : 0.0f;
        for (int i = 0; i < r; ++i) {
          float ai = __shfl(alpha, i, 32);
          bm -= ai * Bs[i][m];
        }
        // s_r = 1 + g . b_r
        float gb = active ? g * bm : 0.0f;
        for (int off = 16; off > 0; off >>= 1) gb += __shfl_xor(gb, off, 32);
        float sr = 1.0f + gb;
        if (active) Bs[r][m] = bm;
        if (lane == 0) ss[r] = sr;
        q += (e / sr) * bm;
      }
      if (active) qs[m] = q;
    }
    __syncthreads();

    if (blk == NBLK - 1) break;   // state update unneeded after final block

    // ---- C[i][j] = sum_r b_r[i] b_r[j] / s_r  (all 256 threads) ----
    {
      const int i = tid >> 4, jj = tid & 15;
      float s = 0.0f;
      for (int r = 0; r < JB; ++r) s += Bs[r][i] * (Bs[r][jj] / ss[r]);
      Cs[i][jj] = s;
    }
    __syncthreads();

    // ---- V = C @ T -> overwrite Xs (current buffer) ; then w += T^T q ----
    {
      const int n0 = wave * 16;
      v8f acc = {};
#pragma unroll
      for (int k = 0; k < JB; k += 4) {
        v2f a, bf;
        a.x  = Cs[l][k + 2 * half + 0];
        a.y  = Cs[l][k + 2 * half + 1];
        bf.x = Ts[k + 2 * half + 0][n0 + l];
        bf.y = Ts[k + 2 * half + 1][n0 + l];
        acc = wmma4(a, bf, acc);
      }
#pragma unroll
      for (int r = 0; r < 8; ++r) Xs[r + 8 * half][n0 + l] = acc[r];
    }
    if (tid < KD) {
      float acc = ws[tid];
      for (int jj = 0; jj < JB; ++jj) acc += Ts[jj][tid] * qs[jj];
      ws[tid] = acc;
    }
    __syncthreads();

    // ---- P -= T^T @ V : wave w owns row panel [16w, 16w+16) ----
    {
      const int m0 = wave * 16;
      for (int nt = 0; nt < 8; ++nt) {
        const int n0 = nt * 16;
        v8f acc;
#pragma unroll
        for (int r = 0; r < 8; ++r) acc[r] = Ps[m0 + r + 8 * half][n0 + l];
#pragma unroll
        for (int k = 0; k < JB; k += 4) {
          v2f a, bf;
          a.x  = -Ts[k + 2 * half + 0][m0 + l];   // A = -(T^T) panel
          a.y  = -Ts[k + 2 * half + 1][m0 + l];
          bf.x = Xs[k + 2 * half + 0][n0 + l];    // Xs holds V
          bf.y = Xs[k + 2 * half + 1][n0 + l];
          acc = wmma4(a, bf, acc);
        }
#pragma unroll
        for (int r = 0; r < 8; ++r) Ps[m0 + r + 8 * half][n0 + l] = acc[r];
      }
    }
    __syncthreads();
  }
}

extern "C" void kernel_launch(void* const* d_in, const int* in_sizes, int n_in,
                              void* d_out, int out_size, void* d_ws, size_t ws_size,
                              hipStream_t stream) {
  const float* data    = (const float*)d_in[0];   // [B, 256, 128] f32
  const float* targets = (const float*)d_in[1];   // [B, 256]      f32
  float*       out     = (float*)d_out;           // [B, 256]      f32
  const int B = in_sizes[1] / NPTS;               // 32
  rls_block_kernel<<<dim3(B), dim3(NTHREADS), 0, stream>>>(data, targets, out);
}